// WorkingMemory_22007412424806
// MI455X (gfx1250) — compile-verified
//
#include <hip/hip_runtime.h>
#include <hip/hip_bf16.h>
#include <math.h>

// Problem constants (match reference)
#define BB     1024   // batch
#define AA     512    // A
#define CC     64     // C
#define DD     10     // D
#define NADDR  1024
#define KCAT   20     // concatenated K = 2*D
#define LOG_TAIL_C 6.2383246250395077847550890931236f  // ln(512)

typedef __attribute__((ext_vector_type(2))) float v2f;
typedef __attribute__((ext_vector_type(4))) float v4f;
typedef __attribute__((ext_vector_type(8))) float v8f;

// ---------------------------------------------------------------------------
// Kernel 1: raw[b,n] = sum_k acat[b,k] * bcat[n,k],  acat = [a0|a1], bcat = [b0|b1]
// One wave32 computes one 16x16 f32 tile via V_WMMA_F32_16X16X4_F32 (5 k-steps).
// grid = 4096 tiles / 8 waves = 512 blocks of 256 threads.
// ---------------------------------------------------------------------------
__global__ __launch_bounds__(256) void attn_raw_wmma(
    const float* __restrict__ a0, const float* __restrict__ a1,
    const float* __restrict__ b0, const float* __restrict__ b1,
    float* __restrict__ raw)
{
  const int wave  = threadIdx.x >> 5;
  const int lane  = threadIdx.x & 31;
  const int tile  = blockIdx.x * 8 + wave;
  const int tilesN = NADDR / 16;            // 64
  const int tm = tile / tilesN;             // row tile (over B)
  const int tn = tile % tilesN;             // col tile (over NADDR)

  const int l16   = lane & 15;
  const int khalf = (lane >> 4) * 2;        // 0 for lanes 0-15, 2 for lanes 16-31
  const int row = tm * 16 + l16;            // A-matrix row (M)
  const int col = tn * 16 + l16;            // B-matrix col (N)

  v8f acc = {};
#pragma unroll
  for (int k0 = 0; k0 < KCAT; k0 += 4) {
    const int ka = k0 + khalf;              // this lane's first K
    v2f av, bv;
    av.x = (ka     < DD) ? a0[row * DD + ka]          : a1[row * DD + (ka - DD)];
    av.y = (ka + 1 < DD) ? a0[row * DD + ka + 1]      : a1[row * DD + (ka + 1 - DD)];
    bv.x = (ka     < DD) ? b0[col * DD + ka]          : b1[col * DD + (ka - DD)];
    bv.y = (ka + 1 < DD) ? b0[col * DD + ka + 1]      : b1[col * DD + (ka + 1 - DD)];
    acc = __builtin_amdgcn_wmma_f32_16x16x4_f32(
        /*neg_a=*/false, av, /*neg_b=*/false, bv,
        /*c_mod=*/(short)0, acc, /*reuse_a=*/false, /*reuse_b=*/false);
  }

  // D layout: VGPR r holds row M = 8*(lane>>4) + r, col N = lane&15
  const int nOut  = tn * 16 + l16;
  const int mBase = tm * 16 + 8 * (lane >> 4);
#pragma unroll
  for (int r = 0; r < 8; ++r) {
    raw[(size_t)(mBase + r) * NADDR + nOut] = acc[r];
  }
}

// ---------------------------------------------------------------------------
// Kernel 2: per-row stats + single streaming pass over m_0/m_1 (256 MB total).
// One block (256 threads) per b.
// ---------------------------------------------------------------------------
__device__ __forceinline__ float logaddexp_f(float x, float y) {
  float mx = fmaxf(x, y);
  float mn = fminf(x, y);
  return mx + log1pf(expf(mn - mx));
}

__device__ __forceinline__ float block_red_max(float v, float* red, int tid) {
  red[tid] = v; __syncthreads();
  for (int s = 128; s > 0; s >>= 1) {
    if (tid < s) red[tid] = fmaxf(red[tid], red[tid + s]);
    __syncthreads();
  }
  float r = red[0]; __syncthreads();
  return r;
}

__device__ __forceinline__ float block_red_sum(float v, float* red, int tid) {
  red[tid] = v; __syncthreads();
  for (int s = 128; s > 0; s >>= 1) {
    if (tid < s) red[tid] = red[tid] + red[tid + s];
    __syncthreads();
  }
  float r = red[0]; __syncthreads();
  return r;
}

__global__ __launch_bounds__(256) void wm_reduce(
    const float* __restrict__ m0, const float* __restrict__ m1,
    const float* __restrict__ v0, const float* __restrict__ v1,
    const float* __restrict__ raw, float* __restrict__ out)
{
  __shared__ float sw[AA];       // attn weights w_i = u_i*(S-u_i)
  __shared__ float red[256];     // reduction scratch
  __shared__ float part[1024];   // 16 groups x 64 cols partials

  const int b   = blockIdx.x;
  const int tid = threadIdx.x;
  const float* rrow = raw + (size_t)b * NADDR;

  // ---- tail logsumexp over raw[b, 512..1023] -> y2 ----
  float t0 = rrow[AA + tid];
  float t1 = rrow[AA + 256 + tid];
  float tmax = block_red_max(fmaxf(t0, t1), red, tid);
  float tsum = block_red_sum(expf(t0 - tmax) + expf(t1 - tmax), red, tid);
  float y2 = tmax + logf(tsum) - LOG_TAIL_C;

  // ---- attn_i = logaddexp(raw_i, y2), i < 512 ----
  float at0 = logaddexp_f(rrow[tid],        y2);
  float at1 = logaddexp_f(rrow[tid + 256],  y2);

  float M = block_red_max(fmaxf(at0, at1), red, tid);
  float u0 = expf(at0 - M), u1 = expf(at1 - M);
  float S  = block_red_sum(u0 + u1, red, tid);
  float Q  = block_red_sum(u0 * u0 + u1 * u1, red, tid);

  sw[tid]       = u0 * (S - u0);
  sw[tid + 256] = u1 * (S - u1);
  __syncthreads();

  // ---- streaming pass over m0,m1: r[b,c] = 2M + log(sum_i w_i e^{m[b,i,c]} + e^{v} Q) ----
  const int c4 = tid & 15;     // float4 column group: columns [c4*4, c4*4+4)
  const int g  = tid >> 4;     // i-group: i in [g*32, (g+1)*32)
  const v4f* p0 = (const v4f*)(m0 + (size_t)b * (AA * CC)) + (size_t)(g * 32) * (CC / 4) + c4;
  const v4f* p1 = (const v4f*)(m1 + (size_t)b * (AA * CC)) + (size_t)(g * 32) * (CC / 4) + c4;

  float acc0[4] = {0.f, 0.f, 0.f, 0.f};
  float acc1[4] = {0.f, 0.f, 0.f, 0.f};

  for (int jj = 0; jj < 32; jj += 8) {
    // lookahead prefetch (gfx1250 global_prefetch_b8); harmless past end (speculative)
    __builtin_prefetch((const void*)(p0 + (size_t)(jj + 16) * (CC / 4)), 0, 0);
    __builtin_prefetch((const void*)(p1 + (size_t)(jj + 16) * (CC / 4)), 0, 0);
#pragma unroll
    for (int j2 = 0; j2 < 8; ++j2) {
      const int j = jj + j2;
      const float w = sw[g * 32 + j];
      v4f x0 = __builtin_nontemporal_load(p0 + (size_t)j * (CC / 4));
      v4f x1 = __builtin_nontemporal_load(p1 + (size_t)j * (CC / 4));
      acc0[0] = fmaf(w, expf(x0.x), acc0[0]);
      acc0[1] = fmaf(w, expf(x0.y), acc0[1]);
      acc0[2] = fmaf(w, expf(x0.z), acc0[2]);
      acc0[3] = fmaf(w, expf(x0.w), acc0[3]);
      acc1[0] = fmaf(w, expf(x1.x), acc1[0]);
      acc1[1] = fmaf(w, expf(x1.y), acc1[1]);
      acc1[2] = fmaf(w, expf(x1.z), acc1[2]);
      acc1[3] = fmaf(w, expf(x1.w), acc1[3]);
    }
  }

  // ---- reduce the 16 i-group partials per column ----
#pragma unroll
  for (int j = 0; j < 4; ++j) part[g * 64 + c4 * 4 + j] = acc0[j];
  __syncthreads();
  float tot0 = 0.f;
  if (tid < 64) {
#pragma unroll
    for (int g2 = 0; g2 < 16; ++g2) tot0 += part[g2 * 64 + tid];
  }
  __syncthreads();
#pragma unroll
  for (int j = 0; j < 4; ++j) part[g * 64 + c4 * 4 + j] = acc1[j];
  __syncthreads();

  if (tid < 64) {
    float tot1 = 0.f;
#pragma unroll
    for (int g2 = 0; g2 < 16; ++g2) tot1 += part[g2 * 64 + tid];
    const float ev0 = expf(v0[b * CC + tid]);
    const float ev1 = expf(v1[b * CC + tid]);
    out[(size_t)b * CC + tid]                  = 2.f * M + logf(tot0 + ev0 * Q);
    out[(size_t)BB * CC + (size_t)b * CC + tid] = 2.f * M + logf(tot1 + ev1 * Q);
  }
}

// ---------------------------------------------------------------------------
// Host-side launcher
// ---------------------------------------------------------------------------
extern "C" void kernel_launch(void* const* d_in, const int* in_sizes, int n_in,
                              void* d_out, int out_size, void* d_ws, size_t ws_size,
                              hipStream_t stream) {
  (void)in_sizes; (void)n_in; (void)out_size; (void)ws_size;
  const float* m0 = (const float*)d_in[0];
  const float* m1 = (const float*)d_in[1];
  const float* a0 = (const float*)d_in[2];
  const float* a1 = (const float*)d_in[3];
  const float* v0 = (const float*)d_in[4];
  const float* v1 = (const float*)d_in[5];
  const float* b0 = (const float*)d_in[6];
  const float* b1 = (const float*)d_in[7];
  float* out = (float*)d_out;
  float* raw = (float*)d_ws;   // 1024*1024 f32 = 4 MB scratch

  // 4096 16x16 tiles, 8 waves/block -> 512 blocks
  hipLaunchKernelGGL(attn_raw_wmma, dim3(512), dim3(256), 0, stream,
                     a0, a1, b0, b1, raw);
  // one block per batch row
  hipLaunchKernelGGL(wm_reduce, dim3(BB), dim3(256), 0, stream,
                     m0, m1, v0, v1, raw, out);
}